// SPDBatchNormImpl_21861383536854
// MI455X (gfx1250) — compile-verified
//
#include <hip/hip_runtime.h>

typedef __attribute__((ext_vector_type(2))) float v2f;
typedef __attribute__((ext_vector_type(8))) float v8f;

constexpr int Bsz = 256;
constexpr int Cch = 40;
constexpr float ETA = 1.0f;
constexpr float EPSC = 1e-5f;

// D-layout (WMMA C/D) tiles: tile(R,C), vgpr v, lane l <-> elem (R*16+v+8*(l>=16), C*16+(l&15))
struct Mat   { v8f t[2][2]; };
// Operand-layout tiles (WMMA A/B form): q[R][C][kk][j], lane l <->
//   elem (R*16 + kk*4 + j + 2*(l>=16), C*16 + (l&15))
struct MatOp { v2f q[2][2][4]; };

// Per-lane constants: lane-half select + diagonal indicator in operand layout.
struct Ctx {
    bool low;
    float d8[8];   // d8[kk*2+j] = 1 if (kk*4 + j + 2*hi == l15)
};

__device__ __forceinline__ Ctx make_ctx(int lane) {
    Ctx cx;
    cx.low = lane < 16;
    int l15 = lane & 15, hi2 = ((lane >> 4) & 1) * 2;
#pragma unroll
    for (int kk = 0; kk < 4; ++kk)
#pragma unroll
        for (int j = 0; j < 2; ++j)
            cx.d8[kk * 2 + j] = (kk * 4 + j + hi2 == l15) ? 1.f : 0.f;
    return cx;
}

// lane <-> lane^16 swap as a VALU op (v_permlanex16_b32 with identity selects)
__device__ __forceinline__ float xor16(float x) {
    unsigned u = __builtin_bit_cast(unsigned, x);
    unsigned r = __builtin_amdgcn_permlanex16(u, u, 0x76543210u, 0xfedcba98u, false, false);
    return __builtin_bit_cast(float, r);
}

// D = A * B, A symmetric (A-operand read from tile(K,I)). Pure WMMA inner loop.
__device__ __forceinline__ void mm_d(Mat &D, const MatOp &A, const MatOp &B) {
#pragma unroll
    for (int I = 0; I < 2; ++I)
#pragma unroll
        for (int J = 0; J < 2; ++J) {
            v8f acc = {0.f, 0.f, 0.f, 0.f, 0.f, 0.f, 0.f, 0.f};
#pragma unroll
            for (int K = 0; K < 2; ++K)
#pragma unroll
                for (int kk = 0; kk < 4; ++kk)
                    acc = __builtin_amdgcn_wmma_f32_16x16x4_f32(
                        false, A.q[K][I][kk], false, B.q[K][J][kk],
                        (short)0, acc, false, false);
            D.t[I][J] = acc;
        }
}

// D-layout -> operand layout, element-wise sel = (low ? direct : xor16(partner)).
// Plain version: O = conv(M)
__device__ __forceinline__ void conv(MatOp &O, const Mat &M, const Ctx &cx) {
    bool low = cx.low;
#pragma unroll
    for (int R = 0; R < 2; ++R)
#pragma unroll
        for (int Cc = 0; Cc < 2; ++Cc) {
            v8f m = M.t[R][Cc];
            float x0 = xor16(m[0]), x1 = xor16(m[1]), x2 = xor16(m[2]), x3 = xor16(m[3]);
            float x4 = xor16(m[4]), x5 = xor16(m[5]), x6 = xor16(m[6]), x7 = xor16(m[7]);
            v2f q0 = {low ? m[0] : x2, low ? m[1] : x3};
            v2f q1 = {low ? m[4] : x6, low ? m[5] : x7};
            v2f q2 = {low ? x0 : m[2], low ? x1 : m[3]};
            v2f q3 = {low ? x4 : m[6], low ? x5 : m[7]};
            O.q[R][Cc][0] = q0; O.q[R][Cc][1] = q1;
            O.q[R][Cc][2] = q2; O.q[R][Cc][3] = q3;
        }
}

// Fused: O = alpha * conv(M) + beta * I   (uses precomputed diag regs, no cmps)
__device__ __forceinline__ void conv_affine(MatOp &O, const Mat &M, float alpha, float beta,
                                            const Ctx &cx) {
    bool low = cx.low;
    float bd[8];
#pragma unroll
    for (int o = 0; o < 8; ++o) bd[o] = beta * cx.d8[o];
#pragma unroll
    for (int R = 0; R < 2; ++R)
#pragma unroll
        for (int Cc = 0; Cc < 2; ++Cc) {
            v8f m = M.t[R][Cc];
            float x0 = xor16(m[0]), x1 = xor16(m[1]), x2 = xor16(m[2]), x3 = xor16(m[3]);
            float x4 = xor16(m[4]), x5 = xor16(m[5]), x6 = xor16(m[6]), x7 = xor16(m[7]);
            float s[8];
            s[0] = low ? m[0] : x2; s[1] = low ? m[1] : x3;
            s[2] = low ? m[4] : x6; s[3] = low ? m[5] : x7;
            s[4] = low ? x0 : m[2]; s[5] = low ? x1 : m[3];
            s[6] = low ? x4 : m[6]; s[7] = low ? x5 : m[7];
#pragma unroll
            for (int kk = 0; kk < 4; ++kk) {
                float a0 = alpha * s[kk * 2 + 0];
                float a1 = alpha * s[kk * 2 + 1];
                if (R == Cc) { a0 += bd[kk * 2 + 0]; a1 += bd[kk * 2 + 1]; }
                v2f q = {a0, a1};
                O.q[R][Cc][kk] = q;
            }
        }
}

// Dst(op) = A*B ; Dst may alias A/B (D temp fully computed before conv).
__device__ __forceinline__ void mmc(MatOp &Dst, const MatOp &A, const MatOp &B, const Ctx &cx) {
    Mat D;
    mm_d(D, A, B);
    conv(Dst, D, cx);
}
// Dst(op) = alpha*(A*B) + beta*I ; alias-safe like mmc.
__device__ __forceinline__ void mmca(MatOp &Dst, const MatOp &A, const MatOp &B,
                                     float alpha, float beta, const Ctx &cx) {
    Mat D;
    mm_d(D, A, B);
    conv_affine(Dst, D, alpha, beta, cx);
}

// ---------- elementwise ops in operand layout ----------
// M = a*X + b*I  (register diag, no per-call compares)
__device__ __forceinline__ void op_axpbI(MatOp &M, float a, const MatOp &X, float b,
                                         const Ctx &cx) {
#pragma unroll
    for (int R = 0; R < 2; ++R)
#pragma unroll
        for (int Cc = 0; Cc < 2; ++Cc)
#pragma unroll
            for (int kk = 0; kk < 4; ++kk)
#pragma unroll
                for (int j = 0; j < 2; ++j) {
                    float v = a * X.q[R][Cc][kk][j];
                    if (R == Cc) v = fmaf(b, cx.d8[kk * 2 + j], v);
                    M.q[R][Cc][kk][j] = v;
                }
}

__device__ __forceinline__ void op_scale(MatOp &M, float a) {
#pragma unroll
    for (int R = 0; R < 2; ++R)
#pragma unroll
        for (int Cc = 0; Cc < 2; ++Cc)
#pragma unroll
            for (int kk = 0; kk < 4; ++kk) M.q[R][Cc][kk] *= a;
}

__device__ __forceinline__ void op_copy(MatOp &D, const MatOp &S) {
#pragma unroll
    for (int R = 0; R < 2; ++R)
#pragma unroll
        for (int Cc = 0; Cc < 2; ++Cc)
#pragma unroll
            for (int kk = 0; kk < 4; ++kk) D.q[R][Cc][kk] = S.q[R][Cc][kk];
}

__device__ __forceinline__ float op_frob(const MatOp &M) {
    float s = 0.f;
#pragma unroll
    for (int R = 0; R < 2; ++R)
#pragma unroll
        for (int Cc = 0; Cc < 2; ++Cc)
#pragma unroll
            for (int kk = 0; kk < 4; ++kk) {
                v2f q = M.q[R][Cc][kk];
                s += q[0] * q[0] + q[1] * q[1];
            }
#pragma unroll
    for (int off = 16; off > 0; off >>= 1) s += __shfl_xor(s, off, 32);
    return sqrtf(fmaxf(s, 1e-30f));
}

// ---------- matrix functions (operand-form in/out, WMMA-based) ----------

// In-place coupled Newton-Schulz sqrt; eig(Y) in (0, ~1.9). First iter peeled (Z0=I).
__device__ __forceinline__ void ns_sqrt_op(MatOp &Y, int iters, const Ctx &cx) {
    MatOp Z, T;
    op_axpbI(Z, -0.5f, Y, 1.5f, cx);      // T0 = 1.5I - 0.5*Y ; Z1 = T0
    mmc(Y, Y, Z, cx);                      // Y1 = Y0*T0
#pragma unroll 1
    for (int it = 1; it < iters; ++it) {
        mmca(T, Z, Y, -0.5f, 1.5f, cx);    // T = 1.5I - 0.5*(Z*Y)   (fused)
        mmc(Y, Y, T, cx);                  // Y = Y*T
        mmc(Z, T, Z, cx);                  // Z = T*Z
    }
}

// Y = A^{1/2}, Z = A^{-1/2} (SPD A), Frobenius-normalized.
__device__ __forceinline__ void ns_invsqrt2_op(const MatOp &A, MatOp &Y, MatOp &Z,
                                               int iters, const Ctx &cx) {
    float c = op_frob(A);
    op_axpbI(Y, 1.f / c, A, 0.f, cx);
    op_axpbI(Z, -0.5f, Y, 1.5f, cx);       // peeled iter 0
    mmc(Y, Y, Z, cx);
    MatOp T;
#pragma unroll 1
    for (int it = 1; it < iters; ++it) {
        mmca(T, Z, Y, -0.5f, 1.5f, cx);
        mmc(Y, Y, T, cx);
        mmc(Z, T, Z, cx);
    }
    op_scale(Y, sqrtf(c));
    op_scale(Z, rsqrtf(c));
}

// S := logm(S), SPD. 5 NS sqrt levels + order-8 Taylor of log(I+z).
__device__ __forceinline__ void logm_op(MatOp &S, const Ctx &cx) {
    float c = op_frob(S);                  // >= lambda_max
    op_scale(S, 1.f / c);                  // eig in (0,1]
    const int iters[5] = {16, 11, 8, 6, 5};
#pragma unroll 1
    for (int lev = 0; lev < 5; ++lev) ns_sqrt_op(S, iters[lev], cx);
    op_axpbI(S, 1.f, S, -1.f, cx);         // z = S^{1/32} - I
    MatOp G;
    op_axpbI(G, -1.f / 8.f, S, 1.f / 7.f, cx);
#pragma unroll 1
    for (int k = 6; k >= 1; --k)
        mmca(G, S, G, -1.f, 1.f / (float)k, cx);   // G = I/k - z*G   (fused, alias-safe)
    MatOp P;
    mmc(P, S, G, cx);                      // log(I+z)
    op_axpbI(S, 32.f, P, logf(c), cx);     // 2^5*log + ln(c)*I
}

// T := expm(T), symmetric. Wave-uniform scaling-and-squaring + Taylor-8.
__device__ __forceinline__ void expm_op(MatOp &T, const Ctx &cx) {
    float a = op_frob(T);
    int j = 0;
    float aa = a;
    while (aa > 0.25f && j < 40) { aa *= 0.5f; ++j; }
    op_scale(T, ldexpf(1.f, -j));
    MatOp G;
    op_axpbI(G, 1.f / 8.f, T, 1.f, cx);
#pragma unroll 1
    for (int k = 7; k >= 1; --k)
        mmca(G, T, G, 1.f / (float)k, 1.f, cx);    // G = I + (T*G)/k  (fused)
    op_copy(T, G);
#pragma unroll 1
    for (int q = 0; q < j; ++q) mmc(T, T, T, cx);  // repeated squaring (alias-safe)
}

// ---------- loads / stores ----------
// row-major 32x32 -> operand layout (direct address map, no shuffles)
__device__ __forceinline__ void load_rm_op(MatOp &M, const float *__restrict__ base, int lane) {
    int l15 = lane & 15, hi2 = ((lane >> 4) & 1) * 2;
#pragma unroll
    for (int R = 0; R < 2; ++R)
#pragma unroll
        for (int Cc = 0; Cc < 2; ++Cc)
#pragma unroll
            for (int kk = 0; kk < 4; ++kk)
#pragma unroll
                for (int j = 0; j < 2; ++j)
                    M.q[R][Cc][kk][j] =
                        base[(R * 16 + kk * 4 + j + hi2) * 32 + Cc * 16 + l15];
}
// D-layout -> row-major (final output only)
__device__ __forceinline__ void store_rm_d(const Mat &M, float *__restrict__ base, int lane) {
    int l15 = lane & 15, hi = (lane >> 4) & 1;
#pragma unroll
    for (int R = 0; R < 2; ++R)
#pragma unroll
        for (int Cc = 0; Cc < 2; ++Cc)
#pragma unroll
            for (int v = 0; v < 8; ++v)
                base[(R * 16 + v + 8 * hi) * 32 + Cc * 16 + l15] = M.t[R][Cc][v];
}
// flat coalesced operand-form storage (layout-preserving)
__device__ __forceinline__ void load_ms_op(MatOp &M, const float *__restrict__ base, int lane) {
#pragma unroll
    for (int R = 0; R < 2; ++R)
#pragma unroll
        for (int Cc = 0; Cc < 2; ++Cc)
#pragma unroll
            for (int kk = 0; kk < 4; ++kk)
#pragma unroll
                for (int j = 0; j < 2; ++j)
                    M.q[R][Cc][kk][j] =
                        base[((((R * 2 + Cc) * 4 + kk) * 2) + j) * 32 + lane];
}
__device__ __forceinline__ void store_ms_op(const MatOp &M, float *__restrict__ base, int lane) {
#pragma unroll
    for (int R = 0; R < 2; ++R)
#pragma unroll
        for (int Cc = 0; Cc < 2; ++Cc)
#pragma unroll
            for (int kk = 0; kk < 4; ++kk)
#pragma unroll
                for (int j = 0; j < 2; ++j)
                    base[((((R * 2 + Cc) * 4 + kk) * 2) + j) * 32 + lane] =
                        M.q[R][Cc][kk][j];
}

// ---------- kernels ----------

// dst[c] = mean_b src[b,c]  (elementwise, layout-agnostic)
__global__ void k_mean_over_batch(const float *__restrict__ src, float *__restrict__ dst) {
    int c = blockIdx.x, tid = threadIdx.x;
#pragma unroll 1
    for (int i = 0; i < 4; ++i) {
        int e = tid + i * 256;
        float s = 0.f;
#pragma unroll 1
        for (int b = 0; b < Bsz; ++b) s += src[(size_t)(b * Cch + c) * 1024 + e];
        dst[(size_t)c * 1024 + e] = s * (1.f / Bsz);
    }
}

// per channel: bm^{1/2}, bm^{-1/2} (stored in operand form)
__global__ void k_bm_invsqrt(const float *__restrict__ bm, float *__restrict__ bm_sq,
                             float *__restrict__ bm_isq) {
    int c = blockIdx.x, lane = threadIdx.x;
    Ctx cx = make_ctx(lane);
    MatOp A, Y, Z;
    load_rm_op(A, bm + (size_t)c * 1024, lane);
    ns_invsqrt2_op(A, Y, Z, 18, cx);
    store_ms_op(Y, bm_sq + (size_t)c * 1024, lane);
    store_ms_op(Z, bm_isq + (size_t)c * 1024, lane);
}

// per matrix: XT = logm(bm^{-1/2} X bm^{-1/2})
__global__ void k_whiten_log(const float *__restrict__ X, const float *__restrict__ bm_isq,
                             float *__restrict__ XT) {
    int w = (int)((blockIdx.x * blockDim.x + threadIdx.x) >> 5);
    int lane = threadIdx.x & 31;
    Ctx cx = make_ctx(lane);
    int c = w % Cch;
    MatOp Xm, Q, S;
    load_rm_op(Xm, X + (size_t)w * 1024, lane);
    load_ms_op(Q, bm_isq + (size_t)c * 1024, lane);
    mmc(S, Xm, Q, cx);           // X*Q      (X symmetric left)
    mmc(S, Q, S, cx);            // Q*X*Q
    logm_op(S, cx);
    store_ms_op(S, XT + (size_t)w * 1024, lane);
}

// per channel: rm geodesic, GT_rm, rm^{-1/2}, Msq = expm(0.5(1-t)logm(rm))
__global__ void k_channel_stats(const float *__restrict__ bm_sq_s, const float *__restrict__ bm_isq_s,
                                const float *__restrict__ GT_s, const float *__restrict__ rmean,
                                const float *__restrict__ tpar,
                                float *__restrict__ GT_rm_s, float *__restrict__ rm_isq_s,
                                float *__restrict__ Msq_s) {
    int c = blockIdx.x, lane = threadIdx.x;
    Ctx cx = make_ctx(lane);

    // bm2 = bm_sq * expm(GT) * bm_sq
    MatOp E1, Bq, bm2;
    load_ms_op(E1, GT_s + (size_t)c * 1024, lane);
    expm_op(E1, cx);
    load_ms_op(Bq, bm_sq_s + (size_t)c * 1024, lane);
    mmc(bm2, E1, Bq, cx);
    mmc(bm2, Bq, bm2, cx);

    // rm = Rsq * expm(ETA*logm(Risq*bm2*Risq)) * Rsq, R = running_mean
    MatOp Rm, Rsq, Risq, S1, rm;
    load_rm_op(Rm, rmean + (size_t)c * 1024, lane);
    ns_invsqrt2_op(Rm, Rsq, Risq, 18, cx);
    mmc(S1, bm2, Risq, cx);
    mmc(S1, Risq, S1, cx);
    logm_op(S1, cx);
    op_scale(S1, ETA);
    expm_op(S1, cx);
    mmc(rm, S1, Rsq, cx);
    mmc(rm, Rsq, rm, cx);

    // GT_rm = logm(bm^{-1/2} rm bm^{-1/2})
    MatOp Qi, S2;
    load_ms_op(Qi, bm_isq_s + (size_t)c * 1024, lane);
    mmc(S2, rm, Qi, cx);
    mmc(S2, Qi, S2, cx);
    logm_op(S2, cx);
    store_ms_op(S2, GT_rm_s + (size_t)c * 1024, lane);

    // rm^{-1/2}
    MatOp Y2, Z2;
    ns_invsqrt2_op(rm, Y2, Z2, 18, cx);
    store_ms_op(Z2, rm_isq_s + (size_t)c * 1024, lane);

    // Msq = expm(0.5*(1-t)*logm(rm))
    logm_op(rm, cx);
    float t = tpar[0];
    op_scale(rm, 0.5f * (1.f - t));
    expm_op(rm, cx);
    store_ms_op(rm, Msq_s + (size_t)c * 1024, lane);
}

// per channel: batch_var -> s = std / sqrt(rv + eps)
__global__ void k_var_s(const float *__restrict__ XT, const float *__restrict__ GT_rm_s,
                        const float *__restrict__ stdv, const float *__restrict__ rvar,
                        float *__restrict__ svec) {
    int c = blockIdx.x, tid = threadIdx.x;
    float acc = 0.f;
#pragma unroll 1
    for (int b = 0; b < Bsz; ++b) {
#pragma unroll
        for (int i = 0; i < 4; ++i) {
            int e = tid + i * 256;
            float d = XT[(size_t)(b * Cch + c) * 1024 + e] - GT_rm_s[(size_t)c * 1024 + e];
            acc += d * d;
        }
    }
    __shared__ float red[256];
    red[tid] = acc;
    __syncthreads();
#pragma unroll
    for (int off = 128; off > 0; off >>= 1) {
        if (tid < off) red[tid] += red[tid + off];
        __syncthreads();
    }
    if (tid == 0) {
        float bv = red[0] * (1.f / Bsz);
        float rv = (1.f - ETA) * rvar[c] + ETA * bv;
        svec[c] = stdv[c] * rsqrtf(rv + EPSC);
    }
}

// per matrix: Xn = Msq * expm(s * logm(rm^{-1/2} X rm^{-1/2})) * Msq
__global__ void k_normalize(const float *__restrict__ X, const float *__restrict__ rm_isq_s,
                            const float *__restrict__ Msq_s, const float *__restrict__ svec,
                            float *__restrict__ out) {
    int w = (int)((blockIdx.x * blockDim.x + threadIdx.x) >> 5);
    int lane = threadIdx.x & 31;
    Ctx cx = make_ctx(lane);
    int c = w % Cch;
    MatOp Xm, Q, S;
    load_rm_op(Xm, X + (size_t)w * 1024, lane);
    load_ms_op(Q, rm_isq_s + (size_t)c * 1024, lane);
    mmc(S, Xm, Q, cx);
    mmc(S, Q, S, cx);
    logm_op(S, cx);
    op_scale(S, svec[c]);            // T = s * logm(...)
    expm_op(S, cx);                  // expm(T)
    MatOp M;
    load_ms_op(M, Msq_s + (size_t)c * 1024, lane);
    mmc(S, S, M, cx);                // expm(T)*Msq
    Mat R2;
    mm_d(R2, M, S);                  // Msq * ...  (final: D-layout)
    store_rm_d(R2, out + (size_t)w * 1024, lane);
}

extern "C" void kernel_launch(void *const *d_in, const int *in_sizes, int n_in,
                              void *d_out, int out_size, void *d_ws, size_t ws_size,
                              hipStream_t stream) {
    const float *X     = (const float *)d_in[0];  // [B,C,32,32]
    const float *stdv  = (const float *)d_in[1];  // [C,1]
    const float *rmean = (const float *)d_in[2];  // [C,32,32]
    const float *rvar  = (const float *)d_in[3];  // [C,1]
    const float *tpar  = (const float *)d_in[4];  // scalar
    float *out = (float *)d_out;
    float *ws = (float *)d_ws;

    float *bm     = ws;              // row-major (mean of X)
    float *bm_sq  = ws + 40960;      // operand-form storage
    float *bm_isq = ws + 81920;
    float *GT     = ws + 122880;
    float *GT_rm  = ws + 163840;
    float *rm_isq = ws + 204800;
    float *Msq    = ws + 245760;
    float *svec   = ws + 286720;

    (void)in_sizes; (void)n_in; (void)out_size; (void)ws_size;

    k_mean_over_batch<<<Cch, 256, 0, stream>>>(X, bm);
    k_bm_invsqrt<<<Cch, 32, 0, stream>>>(bm, bm_sq, bm_isq);
    // XT staged in d_out (operand-form flat); consumed before k_normalize overwrites
    k_whiten_log<<<(Bsz * Cch) / 8, 256, 0, stream>>>(X, bm_isq, out);
    k_mean_over_batch<<<Cch, 256, 0, stream>>>(out, GT);
    k_channel_stats<<<Cch, 32, 0, stream>>>(bm_sq, bm_isq, GT, rmean, tpar,
                                            GT_rm, rm_isq, Msq);
    k_var_s<<<Cch, 256, 0, stream>>>(out, GT_rm, stdv, rvar, svec);
    k_normalize<<<(Bsz * Cch) / 8, 256, 0, stream>>>(X, rm_isq, Msq, svec, out);
}